// MorphologicalFeatureExtractor_19086834663525
// MI455X (gfx1250) — compile-verified
//
#include <hip/hip_runtime.h>
#include <math.h>

typedef float v4f __attribute__((ext_vector_type(4)));
typedef float v2f __attribute__((ext_vector_type(2)));
typedef float v8f __attribute__((ext_vector_type(8)));

#define DIMD 128
#define DIMH 256
#define DIMW 256
#define VOXELS (DIMD * DIMH * DIMW)   // 8,388,608 per batch
#define QUADS  (VOXELS / 4)           // 2,097,152 float4 per batch
#define NBLK   256                    // partial-sum blocks per batch
#define QPB    (QUADS / NBLK)         // 8192 quads per block
#define ITERS  (QPB / 256)            // 32 iterations per thread

// ---------------------------------------------------------------------------
// Pass 1: streaming reduction. 9 linear functionals of the volume.
//   acc[0]=S, [1]=S*d, [2]=S*d^2, [3]=S*h, [4]=S*h^2, [5]=S*w, [6]=S*w^2,
//   [7]=central-box mass, [8]=left-half (w<128) mass
// ---------------------------------------------------------------------------
__global__ __launch_bounds__(256) void morph_partial(const float* __restrict__ seg,
                                                     float* __restrict__ ws)
{
    const int b   = blockIdx.y;
    const int blk = blockIdx.x;
    const int tid = threadIdx.x;

    const v4f* __restrict__ src = (const v4f*)(seg + (size_t)b * VOXELS);

    float acc[9];
    #pragma unroll
    for (int i = 0; i < 9; ++i) acc[i] = 0.f;

    const int base4 = blk * QPB + tid;

    #pragma unroll 4
    for (int j = 0; j < ITERS; ++j) {
        const int idx4 = base4 + j * 256;
        // stream: NT temporal hint — 268MB > L2, don't pollute
        v4f p = __builtin_nontemporal_load(&src[idx4]);

        const int w0 = (idx4 & 63) << 2;       // w of p.x (quads/row = 64)
        const int h  = (idx4 >> 6) & 255;
        const int d  = idx4 >> 14;             // quads/slice = 16384

        const float fd = (float)d;
        const float fh = (float)h;
        const float fw = (float)w0;

        const float s4 = (p.x + p.y) + (p.z + p.w);
        // t1 = sum p_k * k ; t2 = sum p_k * k^2  (k = 0..3 within the quad)
        const float t1 = fmaf(3.f, p.w, fmaf(2.f, p.z, p.y));
        const float t2 = fmaf(9.f, p.w, fmaf(4.f, p.z, p.y));

        acc[0] += s4;
        acc[1]  = fmaf(s4, fd, acc[1]);
        acc[2]  = fmaf(s4 * fd, fd, acc[2]);
        acc[3]  = fmaf(s4, fh, acc[3]);
        acc[4]  = fmaf(s4 * fh, fh, acc[4]);
        // sum p_k*(w0+k)   = s4*w0 + t1
        acc[5] += fmaf(s4, fw, t1);
        // sum p_k*(w0+k)^2 = s4*w0^2 + 2*w0*t1 + t2
        acc[6] += fmaf(fw, fmaf(s4, fw, 2.f * t1), t2);

        // central third box: d in [42,85), h in [85,170), w in [85,170)
        if ((d >= 42) & (d < 85) & (h >= 85) & (h < 170)) {
            float c0 = (w0     >= 85 && w0     < 170) ? p.x : 0.f;
            float c1 = (w0 + 1 >= 85 && w0 + 1 < 170) ? p.y : 0.f;
            float c2 = (w0 + 2 >= 85 && w0 + 2 < 170) ? p.z : 0.f;
            float c3 = (w0 + 3 >= 85 && w0 + 3 < 170) ? p.w : 0.f;
            acc[7] += (c0 + c1) + (c2 + c3);
        }
        // left half: quad is entirely one side (128 % 4 == 0)
        if (w0 < 128) acc[8] += s4;
    }

    // wave32 butterfly reduction
    #pragma unroll
    for (int off = 16; off > 0; off >>= 1) {
        #pragma unroll
        for (int i = 0; i < 9; ++i)
            acc[i] += __shfl_xor(acc[i], off, 32);
    }

    __shared__ float red[8 * 16];
    const int wave = tid >> 5;
    const int lane = tid & 31;
    if (lane == 0) {
        #pragma unroll
        for (int i = 0; i < 9; ++i) red[wave * 16 + i] = acc[i];
        #pragma unroll
        for (int i = 9; i < 16; ++i) red[wave * 16 + i] = 0.f;  // pad for WMMA A
    }
    __syncthreads();

    if (tid < 16) {
        float a = 0.f;
        #pragma unroll
        for (int w = 0; w < 8; ++w) a += red[w * 16 + tid];
        ws[((size_t)b * NBLK + blk) * 16 + tid] = a;
    }
}

// ---------------------------------------------------------------------------
// Pass 2: reduce NBLK partial vectors per batch with v_wmma_f32_16x16x4_f32.
// D += A(16 features x 4 partials) * ones(4x16). Multiplying by 1.0 and
// accumulating in f32 is exact, so the matrix pipe acts as a reduction tree.
// A layout (ISA 7.12.2, 32-bit A 16x4): lanes 0-15 = rows, VGPR0/1 = K0/K1;
// lanes 16-31 = rows, VGPR0/1 = K2/K3. All-ones B is layout-invariant.
// ---------------------------------------------------------------------------
__global__ __launch_bounds__(32) void morph_finalize(const float* __restrict__ ws,
                                                     float* __restrict__ out,
                                                     int nblk)
{
    const int b    = blockIdx.x;
    const int lane = threadIdx.x;                  // 32 lanes, EXEC all ones
    const float* __restrict__ base = ws + (size_t)b * nblk * 16;

    const int m  = lane & 15;                      // feature row
    const int kh = (lane >> 4) << 1;               // K offset: 0 or 2

    v2f ones; ones[0] = 1.f; ones[1] = 1.f;
    v8f c = {0.f, 0.f, 0.f, 0.f, 0.f, 0.f, 0.f, 0.f};

    for (int p = 0; p < nblk; p += 4) {
        v2f a;
        a[0] = base[(p + kh    ) * 16 + m];
        a[1] = base[(p + kh + 1) * 16 + m];
        c = __builtin_amdgcn_wmma_f32_16x16x4_f32(
                /*neg_a=*/false, a, /*neg_b=*/false, ones,
                /*c_mod=*/(short)0, c, /*reuse_a=*/false, /*reuse_b=*/false);
    }

    // D[m][n] identical for all n. C/D layout: VGPR r -> M=r (lanes 0-15),
    // M=r+8 (lanes 16-31). Only features 0..8 are live.
    __shared__ float sums[16];
    if (lane == 0) {
        #pragma unroll
        for (int r = 0; r < 8; ++r) sums[r] = c[r];
    }
    if (lane == 16) {
        sums[8] = c[0];
    }
    __syncthreads();

    if (lane == 0) {
        const float S    = sums[0];
        const float Sd   = sums[1], Sdd = sums[2];
        const float Sh   = sums[3], Shh = sums[4];
        const float Sw   = sums[5], Sww = sums[6];
        const float Scen = sums[7];
        const float Slft = sums[8];

        const float total = fmaxf(S, 1e-6f);
        const float md = Sd / total, mh = Sh / total, mw = Sw / total;
        // var = (E2 - 2*mean*E1 + mean^2*E0) / total  (matches reference exactly)
        const float vd = (Sdd - 2.f * md * Sd + md * md * S) / total;
        const float vh = (Shh - 2.f * mh * Sh + mh * mh * S) / total;
        const float vw = (Sww - 2.f * mw * Sw + mw * mw * S) / total;

        const float vmax = fmaxf(vd, fmaxf(vh, vw));
        const float diam = 2.f * sqrtf(fmaxf(vmax, 0.f));
        const float diam_norm = fminf(fmaxf(diam * 0.01f, 0.f), 1.f);

        const float SPH = (4.0f / 3.0f) * 3.14159f * 1000.0f;  // 10mm sphere vol
        const float count_norm = fminf(fmaxf(S / SPH, 0.f), 5.f) * 0.2f;

        const float vasc = fminf(fmaxf(Scen / total, 0.f), 1.f);

        const float l = Slft, r = S - Slft;
        const float lobe = fminf(fmaxf(fminf(l, r) / fmaxf(fmaxf(l, r), 1e-6f), 0.f), 1.f);

        out[b * 4 + 0] = diam_norm;
        out[b * 4 + 1] = count_norm;
        out[b * 4 + 2] = vasc;
        out[b * 4 + 3] = lobe;
    }
}

extern "C" void kernel_launch(void* const* d_in, const int* in_sizes, int n_in,
                              void* d_out, int out_size, void* d_ws, size_t ws_size,
                              hipStream_t stream)
{
    const float* seg = (const float*)d_in[0];
    float* out = (float*)d_out;
    float* ws  = (float*)d_ws;   // needs B * NBLK * 16 * 4 bytes = 128 KB for B=8

    const int B = in_sizes[0] / VOXELS;

    dim3 grid1(NBLK, B);
    morph_partial<<<grid1, 256, 0, stream>>>(seg, ws);
    morph_finalize<<<B, 32, 0, stream>>>(ws, out, NBLK);
}